// L_GAT_19825569038755
// MI455X (gfx1250) — compile-verified
//
#include <hip/hip_runtime.h>
#include <math.h>

typedef float v2f __attribute__((ext_vector_type(2)));
typedef float v8f __attribute__((ext_vector_type(8)));

#define HIDDEN 128
#define LEAKY 0.01f

// ---- monotone float <-> uint encoding for atomicMax on floats -------------
__device__ __forceinline__ unsigned enc_f32(float f) {
    unsigned u = __float_as_uint(f);
    return (u & 0x80000000u) ? ~u : (u | 0x80000000u);
}
__device__ __forceinline__ float dec_f32(unsigned u) {
    unsigned v = (u & 0x80000000u) ? (u & 0x7FFFFFFFu) : ~u;
    return __uint_as_float(v);
}

// ---- stage 0: zero out[], init segmax=-inf, denom=0 -----------------------
__global__ void k_init(float* __restrict__ out, unsigned* __restrict__ segmax,
                       float* __restrict__ denom, int n, int total) {
    int t = blockIdx.x * blockDim.x + threadIdx.x;
    if (t < total) out[t] = 0.0f;
    if (t < n) {
        segmax[t] = 0x007FFFFFu;   // enc(-inf)
        denom[t]  = 0.0f;
    }
}

// ---- stage 1: ei = x*w_i, ej = x*w_j via v_wmma_f32_16x16x4_f32 -----------
// One wave computes a 16-node tile. B holds w_i in col 0, w_j in col 1.
__global__ void k_logits_wmma(const float* __restrict__ x,
                              const float* __restrict__ wi,
                              const float* __restrict__ wj,
                              float* __restrict__ ei, float* __restrict__ ej,
                              int n) {
    int gtid = blockIdx.x * blockDim.x + threadIdx.x;
    int wave = gtid >> 5;
    int lane = gtid & 31;
    int base = wave * 16;
    if (base >= n) return;                 // wave-uniform: EXEC stays full

    int row = lane & 15;                   // A: M index / B: N (column) index
    int hi  = lane >> 4;                   // 0 -> K pair {0,1}, 1 -> {2,3}

    int nodeA = base + row;
    if (nodeA >= n) nodeA = n - 1;         // clamp (branchless tail-safe load)
    const float* xr = x + (long)nodeA * HIDDEN + hi * 2;

    const float* wsel = (row == 0) ? wi : wj;          // col0=w_i, col1=w_j
    float bmask = (row < 2) ? 1.0f : 0.0f;             // cols 2..15 are zero

    v8f c = {};
    #pragma unroll 4
    for (int k0 = 0; k0 < HIDDEN; k0 += 4) {
        v2f a, b;
        a.x = xr[k0];                       // K = k0 + 2*hi
        a.y = xr[k0 + 1];                   // K = k0 + 2*hi + 1
        b.x = bmask * wsel[k0 + hi * 2];
        b.y = bmask * wsel[k0 + hi * 2 + 1];
        c = __builtin_amdgcn_wmma_f32_16x16x4_f32(
                false, a, false, b, (short)0, c, false, false);
    }

    // D layout: lane (0..15) = column N; VGPR r: lanes0-15 -> M=r, 16-31 -> M=8+r
    if (row < 2) {
        float* dst = (row == 0) ? ei : ej;
        #pragma unroll
        for (int r = 0; r < 8; ++r) {
            int node = base + hi * 8 + r;
            if (node < n) dst[node] = c[r];
        }
    }
}

// ---- stage 2: per-edge logit, leaky-relu, segment max over idx_j ----------
__global__ void k_edge_logit(const int* __restrict__ eidx,
                             const float* __restrict__ ei,
                             const float* __restrict__ ej,
                             float* __restrict__ eval,
                             unsigned* __restrict__ segmax, int E) {
    int e = blockIdx.x * blockDim.x + threadIdx.x;
    if (e >= E) return;
    int j = eidx[e];          // row 0: idx_j (softmax group)
    int i = eidx[E + e];      // row 1: idx_i (aggregation dest)
    float v = ei[i] + ej[j];
    v = (v > 0.0f) ? v : LEAKY * v;
    eval[e] = v;
    atomicMax(&segmax[j], enc_f32(v));
}

// ---- stage 3: exp(e - max) and segment-sum denominator --------------------
__global__ void k_edge_exp(const int* __restrict__ eidx,
                           const unsigned* __restrict__ segmax,
                           float* __restrict__ eval,
                           float* __restrict__ denom, int E) {
    int e = blockIdx.x * blockDim.x + threadIdx.x;
    if (e >= E) return;
    int j = eidx[e];
    float m = dec_f32(segmax[j]);
    float ex = __expf(eval[e] - m);
    eval[e] = ex;
    atomicAdd(&denom[j], ex);
}

// ---- stage 4: scatter out[idx_i] += alpha * x[idx_j]; one wave per edge ---
__global__ void k_scatter(const int* __restrict__ eidx,
                          const float* __restrict__ x,
                          const float* __restrict__ eval,
                          const float* __restrict__ denom,
                          float* __restrict__ out, int E) {
    int gtid = blockIdx.x * blockDim.x + threadIdx.x;
    int wave = gtid >> 5;
    int lane = gtid & 31;
    if (wave >= E) return;
    int j = eidx[wave];
    int i = eidx[E + wave];
    float alpha = eval[wave] / denom[j];
    const float4* xs = (const float4*)(x + (long)j * HIDDEN);
    float*        od = out + (long)i * HIDDEN;
    float4 v = xs[lane];                   // 32 lanes x 16B = full 128-f row
    int c = lane * 4;
    atomicAdd(od + c + 0, alpha * v.x);    // resolves in L2 (x,out L2-resident)
    atomicAdd(od + c + 1, alpha * v.y);
    atomicAdd(od + c + 2, alpha * v.z);
    atomicAdd(od + c + 3, alpha * v.w);
}

// ---- stage 5: relu --------------------------------------------------------
__global__ void k_relu(float* __restrict__ out, int total) {
    int t = blockIdx.x * blockDim.x + threadIdx.x;
    if (t < total) out[t] = fmaxf(out[t], 0.0f);
}

extern "C" void kernel_launch(void* const* d_in, const int* in_sizes, int n_in,
                              void* d_out, int out_size, void* d_ws, size_t ws_size,
                              hipStream_t stream) {
    const float* x    = (const float*)d_in[0];
    const int*   eidx = (const int*)d_in[1];
    const float* wi   = (const float*)d_in[2];
    const float* wj   = (const float*)d_in[3];
    float*       out  = (float*)d_out;

    const int N = in_sizes[0] / HIDDEN;
    const int E = in_sizes[1] / 2;
    const int total = N * HIDDEN;

    // workspace layout (floats): ei[N] | ej[N] | segmax[N](u32) | denom[N] | eval[E]
    float*    ws_f   = (float*)d_ws;
    float*    ei     = ws_f;
    float*    ej     = ws_f + N;
    unsigned* segmax = (unsigned*)(ws_f + 2 * (size_t)N);
    float*    denom  = ws_f + 3 * (size_t)N;
    float*    eval   = ws_f + 4 * (size_t)N;

    const int B = 256;

    k_init<<<(total + B - 1) / B, B, 0, stream>>>(out, segmax, denom, N, total);

    int tiles = (N + 15) / 16;                  // one wave (32 lanes) per tile
    long tthreads = (long)tiles * 32;
    k_logits_wmma<<<(int)((tthreads + B - 1) / B), B, 0, stream>>>(x, wi, wj, ei, ej, N);

    k_edge_logit<<<(E + B - 1) / B, B, 0, stream>>>(eidx, ei, ej, eval, segmax, E);

    k_edge_exp<<<(E + B - 1) / B, B, 0, stream>>>(eidx, segmax, eval, denom, E);

    long sthreads = (long)E * 32;               // one wave per edge
    k_scatter<<<(int)((sthreads + B - 1) / B), B, 0, stream>>>(eidx, x, eval, denom, out, E);

    k_relu<<<(total + B - 1) / B, B, 0, stream>>>(out, total);
}